// TestQKNormRopeModelNoBias_76587856822535
// MI455X (gfx1250) — compile-verified
//
#include <hip/hip_runtime.h>
#include <hip/hip_bf16.h>

typedef __attribute__((ext_vector_type(4))) float v4f;   // native vector -> b128 vmem

namespace {
constexpr int   SEQ         = 8192;
constexpr int   NHEADS      = 32;
constexpr int   NKV         = 8;
constexpr int   HD          = 128;
constexpr int   ROW_IN      = NHEADS * HD + 2 * NKV * HD;   // 6144 floats per qkv row
constexpr int   K_OFF       = NHEADS * HD;                  // 4096
constexpr int   V_OFF       = NHEADS * HD + NKV * HD;       // 5120
constexpr int   HEADS_TOTAL = NHEADS + 2 * NKV;             // 48 head-tiles per row
constexpr float EPS         = 1e-6f;
}

// One wave32 per (row, head) tile of 128 f32: 4 floats (one b128) per lane.
// RMS-norm reduction = 5-step wave32 butterfly; RoPE partner (dim +-64) is
// exactly lane^16, one cross-lane shuffle. Streaming data uses TH=NT.
__global__ __launch_bounds__(256)
void qknorm_rope_kernel(const float* __restrict__ qkv,
                        const float* __restrict__ qw,
                        const float* __restrict__ kw,
                        const float* __restrict__ cw,
                        const float* __restrict__ sw,
                        float* __restrict__ out)
{
    const int lane = threadIdx.x & 31;
    const int wave = (blockIdx.x * blockDim.x + threadIdx.x) >> 5;
    const int row  = wave / HEADS_TOTAL;
    const int head = wave - row * HEADS_TOTAL;          // uniform across the wave
    if (row >= SEQ) return;

    const int e = lane << 2;                            // element offset in head dim

    float* const qOut = out;                                        // (seq,32,128)
    float* const kOut = out  + (size_t)SEQ * (NHEADS * HD);         // (seq, 8,128)
    float* const vOut = kOut + (size_t)SEQ * (NKV * HD);            // (seq, 8,128)

    if (head >= NHEADS + NKV) {                         // ---- V: pure NT copy ----
        const int h = head - (NHEADS + NKV);
        const v4f v = __builtin_nontemporal_load(
            reinterpret_cast<const v4f*>(qkv + (size_t)row * ROW_IN + V_OFF + h * HD + e));
        __builtin_nontemporal_store(v,
            reinterpret_cast<v4f*>(vOut + (size_t)row * (NKV * HD) + h * HD + e));
        return;
    }

    const bool isQ  = head < NHEADS;
    const int  h    = isQ ? head : head - NHEADS;
    const float* src  = qkv + (size_t)row * ROW_IN + (isQ ? h * HD : K_OFF + h * HD) + e;
    float*       dst  = isQ ? qOut + (size_t)row * (NHEADS * HD) + h * HD + e
                            : kOut + (size_t)row * (NKV * HD)    + h * HD + e;
    const float* wptr = (isQ ? qw : kw) + e;

    const v4f x = __builtin_nontemporal_load(reinterpret_cast<const v4f*>(src));

    // ---- RMS norm: sum of squares over 128 elements (wave32 butterfly) ----
    float ss = x.x * x.x + x.y * x.y + x.z * x.z + x.w * x.w;
#pragma unroll
    for (int m = 16; m >= 1; m >>= 1)
        ss += __shfl_xor(ss, m, 32);
    const float inv = rsqrtf(ss * (1.0f / HD) + EPS);

    const v4f w = *reinterpret_cast<const v4f*>(wptr);   // WGP$-resident (2 KB total)
    v4f n;
    n.x = x.x * inv * w.x;
    n.y = x.y * inv * w.y;
    n.z = x.z * inv * w.z;
    n.w = x.w * inv * w.w;

    // ---- RoPE: rotate_half partner lives in lane^16 (dim offset +-64) ----
    v4f p;
    p.x = __shfl_xor(n.x, 16, 32);
    p.y = __shfl_xor(n.y, 16, 32);
    p.z = __shfl_xor(n.z, 16, 32);
    p.w = __shfl_xor(n.w, 16, 32);
    const float sg = (lane < 16) ? -1.0f : 1.0f;   // -x2 for first half, +x1 for second

    const v4f c = *reinterpret_cast<const v4f*>(cw + e);
    const v4f s = *reinterpret_cast<const v4f*>(sw + e);
    v4f o;
    o.x = fmaf(sg * p.x, s.x, n.x * c.x);
    o.y = fmaf(sg * p.y, s.y, n.y * c.y);
    o.z = fmaf(sg * p.z, s.z, n.z * c.z);
    o.w = fmaf(sg * p.w, s.w, n.w * c.w);

    __builtin_nontemporal_store(o, reinterpret_cast<v4f*>(dst));
}

extern "C" void kernel_launch(void* const* d_in, const int* in_sizes, int n_in,
                              void* d_out, int out_size, void* d_ws, size_t ws_size,
                              hipStream_t stream) {
    (void)in_sizes; (void)n_in; (void)out_size; (void)d_ws; (void)ws_size;
    const float* qkv = (const float*)d_in[0];
    const float* qw  = (const float*)d_in[1];
    const float* kw  = (const float*)d_in[2];
    const float* cw  = (const float*)d_in[3];
    const float* sw  = (const float*)d_in[4];
    float* out = (float*)d_out;

    constexpr int tiles          = SEQ * HEADS_TOTAL;   // 393216 wave-tiles
    constexpr int threadsPerBlk  = 256;                 // 8 waves per block
    constexpr int wavesPerBlk    = threadsPerBlk / 32;
    constexpr int blocks         = (tiles + wavesPerBlk - 1) / wavesPerBlk;  // 49152

    qknorm_rope_kernel<<<blocks, threadsPerBlk, 0, stream>>>(qkv, qw, kw, cw, sw, out);
}